// HNetMax_42271068127507
// MI455X (gfx1250) — compile-verified
//
#include <hip/hip_runtime.h>
#include <hip/hip_bf16.h>

// ---------------------------------------------------------------------------
// Masked column-max: out[p,c] = max_{n : states[n,c]!=0} x[p,n]  (0.0 if none)
//   x:      [2048, 8192] f32
//   states: [8192, 1024] i32
//   out:    [2048, 1024] f32
//
// Max-plus semiring "GEMM" -> WMMA inapplicable (sum-of-products only), so
// this is VALU-bound: ~17.2G compare-selects. Inner loop runs at 1.5 VALU ops
// per (p,c,n) via node-pairing into v_max3_num_f32, with bit-packed mask
// (32x traffic cut) and CDNA5 TDM (tensor_load_to_lds) double-buffered async
// staging of x/mask tiles into LDS (TENSORcnt pipeline).
// ---------------------------------------------------------------------------

#define N_PTS   2048
#define N_NODES 8192
#define N_CMP   1024
#define CW      (N_CMP / 32)     // 32 mask words per node row

// main-kernel tiling
#define TP 128                   // points per block
#define TC 64                    // columns per block
#define KN 128                   // node chunk
#define NCHUNK (N_NODES / KN)    // 64

#define XB   (TP * KN * 4)                 // 65536 B  x tile
#define MB_  (KN * 2 * 4)                  // 1024  B  mask tile (2 words/node)
#define BUFB (XB + MB_)                    // 66560 B per buffer
#define SMEM_TOTAL (2 * BUFB)              // 133120 B (double buffered)

typedef unsigned int v4u __attribute__((ext_vector_type(4)));
typedef int          v4i __attribute__((ext_vector_type(4)));
typedef int          v8i __attribute__((ext_vector_type(8)));

#define NEGINF (-__builtin_inff())

// ---------------------------------------------------------------------------
// TDM: issue a 2D tile DMA global->LDS.  D# bit layout per CDNA5 ISA ch.8:
//  group0: [1:0]=count=1, [63:32]=lds_addr, [120:64]=global_addr, [127:126]=2
//  group1: [17:16]=data_size(2 => 4B), [79:48]=tensor_dim0, [111:80]=tensor_dim1,
//          [127:112]=tile_dim0, [143:128]=tile_dim1, [207:160]=tensor_dim0_stride
// Tiles are always in-bounds here, tensor dims only need to cover the tile.
// ---------------------------------------------------------------------------
__device__ __forceinline__ void tdm_tile_load_2d(unsigned lds_off,
                                                 const void* gptr,
                                                 unsigned tile_d0, unsigned tile_d1,
                                                 unsigned long long stride0_elems,
                                                 unsigned tensor_d0, unsigned tensor_d1)
{
    unsigned long long ga = (unsigned long long)gptr;
    v4u g0;
    g0.x = 1u;                                   // count=1, user mode
    g0.y = lds_off;                              // LDS byte address
    g0.z = (unsigned)ga;                         // global_addr[31:0]
    g0.w = (unsigned)((ga >> 32) & 0x01FFFFFFull) | (2u << 30);  // addr[56:32] | type=2

    v8i g1;
    g1[0] = (int)(2u << 16);                                         // data_size = 4B
    g1[1] = (int)((tensor_d0 & 0xFFFFu) << 16);                      // tensor_dim0[15:0]
    g1[2] = (int)(((tensor_d0 >> 16) & 0xFFFFu) | ((tensor_d1 & 0xFFFFu) << 16));
    g1[3] = (int)(((tensor_d1 >> 16) & 0xFFFFu) | (tile_d0 << 16));  // tile_dim0
    g1[4] = (int)(tile_d1 & 0xFFFFu);                                // tile_dim1 (tile_dim2=0)
    g1[5] = (int)(unsigned)(stride0_elems & 0xFFFFFFFFull);          // dim0_stride[31:0]
    g1[6] = (int)(unsigned)((stride0_elems >> 32) & 0xFFFFull);      // dim0_stride[47:32]
    g1[7] = 0;

    v4i z4 = {0, 0, 0, 0};                       // groups 2/3: 2D tensor -> unused
#if defined(__clang_major__) && (__clang_major__ >= 23)
    v8i z8 = {0, 0, 0, 0, 0, 0, 0, 0};
    __builtin_amdgcn_tensor_load_to_lds(g0, g1, z4, z4, z8, 0);
#else
    __builtin_amdgcn_tensor_load_to_lds(g0, g1, z4, z4, 0);
#endif
}

// ---------------------------------------------------------------------------
// Main kernel (placed first so the disasm snippet shows the TDM codegen):
// tiled masked col-max with TDM double buffering.
// Block: 256 threads = 16 col-groups (tx) x 16 point-groups (ty).
// Thread owns 8 points x 4 columns of accumulators.
// ---------------------------------------------------------------------------
__global__ void __launch_bounds__(256, 1)
hnetmax_main(const float* __restrict__ x,
             const unsigned* __restrict__ mb,
             float* __restrict__ out)
{
    extern __shared__ char smem[];

    const int tx = threadIdx.x & 15;     // column group: cols c0 + tx*4 .. +3
    const int ty = threadIdx.x >> 4;     // point group : pts  p0 + ty*8 .. +7
    const int c0 = blockIdx.x * TC;
    const int p0 = blockIdx.y * TP;
    const int cw0   = c0 >> 5;           // first mask word of this col tile
    const int wsel  = tx >> 3;           // which of the 2 tile words
    const unsigned shift = (unsigned)((tx & 7) * 4);

    float acc[8][4];
#pragma unroll
    for (int i = 0; i < 8; ++i)
#pragma unroll
        for (int j = 0; j < 4; ++j) acc[i][j] = NEGINF;

    const bool issuer = (threadIdx.x < 32);   // wave 0 drives the TDM

    // ---- prologue: DMA chunk 0 into buffer 0 ----
    if (issuer) {
        tdm_tile_load_2d((unsigned)(size_t)(smem),
                         x + (size_t)p0 * N_NODES,           // x tile [TP][KN]
                         KN, TP, (unsigned long long)N_NODES,
                         N_NODES, N_PTS);
        tdm_tile_load_2d((unsigned)(size_t)(smem + XB),
                         mb + cw0,                           // mask tile [KN][2]
                         2, KN, (unsigned long long)CW,
                         CW, N_NODES);
    }

    for (int ch = 0; ch < NCHUNK; ++ch) {
        const int buf = ch & 1;

        if (issuer) {
            if (ch + 1 < NCHUNK) {
                const char* nb = smem + (buf ^ 1) * BUFB;
                const int k1 = (ch + 1) * KN;
                tdm_tile_load_2d((unsigned)(size_t)nb,
                                 x + (size_t)p0 * N_NODES + k1,
                                 KN, TP, (unsigned long long)N_NODES,
                                 N_NODES, N_PTS);
                tdm_tile_load_2d((unsigned)(size_t)(nb + XB),
                                 mb + (size_t)k1 * CW + cw0,
                                 2, KN, (unsigned long long)CW,
                                 CW, N_NODES);
                __builtin_amdgcn_s_wait_tensorcnt(2);  // chunk ch fully landed
            } else {
                __builtin_amdgcn_s_wait_tensorcnt(0);
            }
        }
        __syncthreads();                                // data visible to all waves

        const float*    xs = (const float*)(smem + buf * BUFB);
        const unsigned* ms = (const unsigned*)(smem + buf * BUFB + XB);

        for (int kk = 0; kk < KN; kk += 4) {
            __align__(16) float xv[8][4];
#pragma unroll
            for (int i = 0; i < 8; ++i)
                *(float4*)&xv[i][0] =
                    *(const float4*)&xs[(ty * 8 + i) * KN + kk];

            unsigned mwv[4];
#pragma unroll
            for (int j = 0; j < 4; ++j)
                mwv[j] = ms[(kk + j) * 2 + wsel] >> shift;

            // Node-paired: 2x v_cndmask + 1x v_max3_num_f32 per 2 node-elems
            // => 1.5 VALU ops per (p,c,n).
#pragma unroll
            for (int j = 0; j < 4; j += 2) {
                const unsigned w0 = mwv[j];
                const unsigned w1 = mwv[j + 1];
#pragma unroll
                for (int cj = 0; cj < 4; ++cj) {
                    const bool on0 = (w0 & (1u << cj)) != 0u;  // v_cmp vs const
                    const bool on1 = (w1 & (1u << cj)) != 0u;
#pragma unroll
                    for (int i = 0; i < 8; ++i) {
                        float a = on0 ? xv[i][j]     : NEGINF; // v_cndmask
                        float b = on1 ? xv[i][j + 1] : NEGINF; // v_cndmask
                        acc[i][cj] = fmaxf(fmaxf(acc[i][cj], a), b); // v_max3
                    }
                }
            }
        }
        __syncthreads();      // protect buf before it is overwritten by TDM
    }

    // ---- epilogue: -inf (no active node in column) -> 0.0, vector store ----
#pragma unroll
    for (int i = 0; i < 8; ++i) {
        float4 o;
        o.x = (acc[i][0] == NEGINF) ? 0.0f : acc[i][0];
        o.y = (acc[i][1] == NEGINF) ? 0.0f : acc[i][1];
        o.z = (acc[i][2] == NEGINF) ? 0.0f : acc[i][2];
        o.w = (acc[i][3] == NEGINF) ? 0.0f : acc[i][3];
        *(float4*)&out[(size_t)(p0 + ty * 8 + i) * N_CMP + c0 + tx * 4] = o;
    }
}

// ---------------------------------------------------------------------------
// Kernel 1: pack mask bits.  maskbits[n][w] bit b = (states[n][w*32+b] != 0)
// ---------------------------------------------------------------------------
__global__ void hnetmax_pack_mask(const int* __restrict__ st,
                                  unsigned* __restrict__ mb)
{
    int t = blockIdx.x * blockDim.x + threadIdx.x;     // 0 .. N_NODES*CW-1
    int n = t >> 5;
    int w = t & 31;
    const int4* p = (const int4*)(st + (size_t)n * N_CMP + w * 32);
    unsigned bits = 0u;
#pragma unroll
    for (int q = 0; q < 8; ++q) {
        int4 v = p[q];
        if (v.x) bits |= 1u << (q * 4 + 0);
        if (v.y) bits |= 1u << (q * 4 + 1);
        if (v.z) bits |= 1u << (q * 4 + 2);
        if (v.w) bits |= 1u << (q * 4 + 3);
    }
    mb[t] = bits;
}

// ---------------------------------------------------------------------------
// Fallback (only if workspace is too small to hold the 1 MB packed mask).
// ---------------------------------------------------------------------------
__global__ void hnetmax_naive(const float* __restrict__ x,
                              const int* __restrict__ st,
                              float* __restrict__ out)
{
    int c = blockIdx.x * blockDim.x + threadIdx.x;
    int p = blockIdx.y;
    float acc = NEGINF;
    for (int n = 0; n < N_NODES; ++n)
        if (st[(size_t)n * N_CMP + c])
            acc = fmaxf(acc, x[(size_t)p * N_NODES + n]);
    out[(size_t)p * N_CMP + c] = (acc == NEGINF) ? 0.0f : acc;
}

// ---------------------------------------------------------------------------
extern "C" void kernel_launch(void* const* d_in, const int* in_sizes, int n_in,
                              void* d_out, int out_size, void* d_ws, size_t ws_size,
                              hipStream_t stream)
{
    const float* x  = (const float*)d_in[0];
    const int*   st = (const int*)d_in[1];
    float*       out = (float*)d_out;

    const size_t need = (size_t)N_NODES * CW * sizeof(unsigned);   // 1 MB
    if (ws_size < need) {
        hnetmax_naive<<<dim3(N_CMP / 256, N_PTS), 256, 0, stream>>>(x, st, out);
        return;
    }

    unsigned* mb = (unsigned*)d_ws;
    hnetmax_pack_mask<<<(N_NODES * CW) / 256, 256, 0, stream>>>(st, mb);

    (void)hipFuncSetAttribute(reinterpret_cast<const void*>(hnetmax_main),
                              hipFuncAttributeMaxDynamicSharedMemorySize,
                              (int)SMEM_TOTAL);
    hnetmax_main<<<dim3(N_CMP / TC, N_PTS / TP), 256, SMEM_TOTAL, stream>>>(x, mb, out);
}